// MacroDensityUniformityOp_77781857731240
// MI455X (gfx1250) — compile-verified
//
#include <hip/hip_runtime.h>
#include <hip/hip_bf16.h>

typedef __attribute__((ext_vector_type(2))) float v2f;
typedef __attribute__((ext_vector_type(8))) float v8f;

#define BXC   32
#define BYC   32
#define NBINS (BXC * BYC)
// clip upper bound: BX - 1 - EPS = 31 - 1e-6
#define FCLIP (31.0f - 1e-6f)

// ---------------------------------------------------------------------------
// Zero the 1024-bin global accumulator (lives in d_ws).
// ---------------------------------------------------------------------------
__global__ void zero_bins_kernel(float* __restrict__ density) {
    int i = blockIdx.x * blockDim.x + threadIdx.x;
    if (i < NBINS) density[i] = 0.0f;
}

// ---------------------------------------------------------------------------
// Bilinear deposit of one point into the LDS-private histogram.
// ds_add_f32 atomics — 64 banks, 1024 bins, low contention.
// ---------------------------------------------------------------------------
__device__ __forceinline__ void deposit(float* __restrict__ lds,
                                        float x, float y, float area) {
    float fx = fminf(fmaxf(x * 32.0f, 0.0f), FCLIP);
    float fy = fminf(fmaxf(y * 32.0f, 0.0f), FCLIP);
    int   ix = (int)fx;
    int   iy = (int)fy;
    float dx = fx - (float)ix;
    float dy = fy - (float)iy;
    int   ix1 = min(ix + 1, BXC - 1);
    int   iy1 = min(iy + 1, BYC - 1);
    float mdx = 1.0f - dx;
    float mdy = 1.0f - dy;
    atomicAdd(&lds[iy  * BXC + ix ], mdx * mdy * area);
    atomicAdd(&lds[iy  * BXC + ix1], dx  * mdy * area);
    atomicAdd(&lds[iy1 * BXC + ix ], mdx * dy  * area);
    atomicAdd(&lds[iy1 * BXC + ix1], dx  * dy  * area);
}

// ---------------------------------------------------------------------------
// Main scatter kernel: grid-stride float4 streaming loads, LDS-private bins,
// global float-atomic merge at the end.
// ---------------------------------------------------------------------------
__global__ void hist_kernel(const float* __restrict__ pos,
                            const unsigned char* __restrict__ mask,
                            const float* __restrict__ msx,
                            const float* __restrict__ msy,
                            float* __restrict__ density,
                            int n_mov, int n_phys) {
    __shared__ float lds[NBINS];
    for (int i = threadIdx.x; i < NBINS; i += blockDim.x) lds[i] = 0.0f;
    __syncthreads();

    const float* __restrict__ xs = pos;
    const float* __restrict__ ys = pos + n_phys;

    const int tid      = blockIdx.x * blockDim.x + threadIdx.x;
    const int nthreads = gridDim.x * blockDim.x;
    const int nvec     = n_mov >> 2;   // 10,000,000 is divisible by 4

    for (int i = tid; i < nvec; i += nthreads) {
        float4   x4 = ((const float4*)xs)[i];
        float4   y4 = ((const float4*)ys)[i];
        float4   sx = ((const float4*)msx)[i];
        float4   sy = ((const float4*)msy)[i];
        unsigned m4 = ((const unsigned*)mask)[i];

        float a0 = fminf(sx.x * 32.0f, 1.0f) * fminf(sy.x * 32.0f, 1.0f)
                 * (((m4      ) & 0xFFu) ? 1.0f : 0.0f);
        float a1 = fminf(sx.y * 32.0f, 1.0f) * fminf(sy.y * 32.0f, 1.0f)
                 * (((m4 >>  8) & 0xFFu) ? 1.0f : 0.0f);
        float a2 = fminf(sx.z * 32.0f, 1.0f) * fminf(sy.z * 32.0f, 1.0f)
                 * (((m4 >> 16) & 0xFFu) ? 1.0f : 0.0f);
        float a3 = fminf(sx.w * 32.0f, 1.0f) * fminf(sy.w * 32.0f, 1.0f)
                 * (((m4 >> 24) & 0xFFu) ? 1.0f : 0.0f);

        deposit(lds, x4.x, y4.x, a0);
        deposit(lds, x4.y, y4.y, a1);
        deposit(lds, x4.z, y4.z, a2);
        deposit(lds, x4.w, y4.w, a3);
    }
    // scalar tail (defensive; empty for n_mov % 4 == 0)
    for (int i = (nvec << 2) + tid; i < n_mov; i += nthreads) {
        float a = fminf(msx[i] * 32.0f, 1.0f) * fminf(msy[i] * 32.0f, 1.0f)
                * (mask[i] ? 1.0f : 0.0f);
        deposit(lds, xs[i], ys[i], a);
    }

    __syncthreads();
    for (int i = threadIdx.x; i < NBINS; i += blockDim.x) {
        unsafeAtomicAdd(&density[i], lds[i]);   // native global_atomic_add_f32
    }
}

// ---------------------------------------------------------------------------
// Cross-wave sum of one f32 value per lane using v_wmma_f32_16x16x4_f32.
// A (16x4, documented layout): lane L<16 holds A[L][0]=s, A[L][1]=0;
//                              lane L>=16 holds A[L-16][2]=s, A[L-16][3]=0.
// B = all-ones (layout independent).  D[m][n] = s_m + s_{m+16}, replicated
// over columns.  Sum of the 8 C/D VGPRs gives rows 0-7 (lanes 0-15) or rows
// 8-15 (lanes 16-31); adding the xor-16 partner yields the full 32-lane sum
// on every lane.  Requires EXEC all-ones: launched as one full wave of 32.
// ---------------------------------------------------------------------------
__device__ __forceinline__ float wave_sum_wmma(float s) {
    v2f a;  a.x = s;    a.y = 0.0f;
    v2f b;  b.x = 1.0f; b.y = 1.0f;
    v8f c = {};
    c = __builtin_amdgcn_wmma_f32_16x16x4_f32(
            /*neg_a=*/false, a, /*neg_b=*/false, b,
            /*c_mod=*/(short)0, c, /*reuse_a=*/false, /*reuse_b=*/false);
    float p = c[0] + c[1] + c[2] + c[3] + c[4] + c[5] + c[6] + c[7];
    return p + __shfl_xor(p, 16, 32);
}

// ---------------------------------------------------------------------------
// Variance (ddof=1) over the 1024 bins, single wave32, two-pass for accuracy.
// ---------------------------------------------------------------------------
__global__ void var_kernel(const float* __restrict__ density,
                           float* __restrict__ out) {
    const int lane = threadIdx.x;           // 0..31, one full wave
    float v[32];
#pragma unroll
    for (int i = 0; i < 32; ++i) v[i] = density[lane + 32 * i];

    float s = 0.0f;
#pragma unroll
    for (int i = 0; i < 32; ++i) s += v[i];
    float total = wave_sum_wmma(s);
    float mean  = total * (1.0f / (float)NBINS);

    float q = 0.0f;
#pragma unroll
    for (int i = 0; i < 32; ++i) { float d = v[i] - mean; q += d * d; }
    float ssq = wave_sum_wmma(q);

    if (lane == 0) out[0] = ssq * (1.0f / (float)(NBINS - 1));
}

// ---------------------------------------------------------------------------
// Host launcher.
// d_in[0]=pos (2*NUM_PHYS f32), d_in[1]=macro_mask (u8), d_in[2]=macro_size_x,
// d_in[3]=macro_size_y.  d_out = 1 f32 (variance).  d_ws holds 1024-bin accum.
// ---------------------------------------------------------------------------
extern "C" void kernel_launch(void* const* d_in, const int* in_sizes, int n_in,
                              void* d_out, int out_size, void* d_ws, size_t ws_size,
                              hipStream_t stream) {
    const float*         pos  = (const float*)d_in[0];
    const unsigned char* mask = (const unsigned char*)d_in[1];
    const float*         msx  = (const float*)d_in[2];
    const float*         msy  = (const float*)d_in[3];

    const int n_mov  = in_sizes[2];        // 10,000,000
    const int n_phys = in_sizes[0] / 2;    // 11,000,000

    float* density = (float*)d_ws;         // 1024 floats of scratch
    float* out     = (float*)d_out;

    zero_bins_kernel<<<(NBINS + 255) / 256, 256, 0, stream>>>(density);

    // 1024 blocks x 256 threads (8 wave32 each): ~9.5 float4 iterations per
    // thread; enough waves in flight to saturate HBM, merge traffic is only
    // 1M global float atomics spread over 1024 addresses.
    hist_kernel<<<1024, 256, 0, stream>>>(pos, mask, msx, msy, density,
                                          n_mov, n_phys);

    var_kernel<<<1, 32, 0, stream>>>(density, out);
}